// DGCNN_53601191854514
// MI455X (gfx1250) — compile-verified
//
#include <hip/hip_runtime.h>
#include <hip/hip_bf16.h>
#include <cfloat>

// ---------------------------------------------------------------------------
// DGCNN (3x EdgeConv, K=20, N=16384) for gfx1250 (MI455X), f32 WMMA 16x16x4.
// Round 2: branchless hot loops via channel-padded inputs/weights.
// ---------------------------------------------------------------------------

typedef float v2f __attribute__((ext_vector_type(2)));
typedef float v8f __attribute__((ext_vector_type(8)));

#define NPTS      16384
#define KNN_K     20
#define NEG_SLOPE 0.2f

static __device__ __forceinline__ v8f wmma_f32(v2f a, v2f b, v8f c) {
    // D = A(16x4 f32) * B(4x16 f32) + C(16x16 f32)
    return __builtin_amdgcn_wmma_f32_16x16x4_f32(
        /*neg_a=*/false, a, /*neg_b=*/false, b,
        /*c_mod=*/(short)0, c, /*reuse_a=*/false, /*reuse_b=*/false);
}

static __device__ __forceinline__ v8f v8f_zero() {
    v8f z = {0.f, 0.f, 0.f, 0.f, 0.f, 0.f, 0.f, 0.f};
    return z;
}

// ---------------------------------------------------------------------------
// Setup: pad pos (N x 3) to stride-4 rows with channel 3 = 0.
// ---------------------------------------------------------------------------
__global__ void pad_pos_kernel(const float* __restrict__ pos, float* __restrict__ xpad, int n) {
    int t = blockIdx.x * blockDim.x + threadIdx.x;
    if (t < n * 4) {
        int i = t >> 2, c = t & 3;
        xpad[t] = (c < 3) ? pos[i * 3 + c] : 0.f;
    }
}

// Setup: pad W1 of EdgeConv-0 (6 x 64) to (8 x 64) with zero rows.
__global__ void pad_w1_kernel(const float* __restrict__ w, float* __restrict__ wp) {
    int t = blockIdx.x * blockDim.x + threadIdx.x;  // 512 entries
    if (t < 8 * 64) {
        int k = t >> 6, c = t & 63;
        wp[t] = (k < 6) ? w[k * 64 + c] : 0.f;
    }
}

// ---------------------------------------------------------------------------
// Squared norms: sq[i] = sum_c x[i][c]^2   (CIN = row stride, pad cols are 0)
// ---------------------------------------------------------------------------
template <int CIN>
__global__ void sqnorm_kernel(const float* __restrict__ x, float* __restrict__ sq, int n) {
    int t = blockIdx.x * blockDim.x + threadIdx.x;
    if (t < n) {
        float s = 0.f;
#pragma unroll
        for (int c = 0; c < CIN; ++c) {
            float v = x[t * CIN + c];
            s += v * v;
        }
        sq[t] = s;
    }
}

// ---------------------------------------------------------------------------
// KNN: one wave handles 16 query points (WMMA N dim). Candidates stream as
// 16-row A tiles; dot(cand_m, query_n) accumulated with v_wmma_f32_16x16x4_f32.
// CIN must be a multiple of 4 (inputs pre-padded) -> branchless b64 fetches.
//
// 32-bit operand striping (ISA 7.12.2):
//   A (16x4): lane L -> row M=L&15 ; K = 4*kc + (L<16?0:2) (+1 in 2nd VGPR)
//   B (4x16): lane L -> col N=L&15 ; same K striping
//   D (16x16): lane L, vgpr v -> row M = v + (L<16?0:8), col N = L&15
// ---------------------------------------------------------------------------
template <int CIN>
__global__ void __launch_bounds__(32) knn_kernel(const float* __restrict__ x,
                                                 const float* __restrict__ sq,
                                                 int* __restrict__ knn_idx, int n) {
    static_assert(CIN % 4 == 0, "CIN must be padded to a multiple of 4");
    constexpr int KC = CIN / 4;
    const int lane  = threadIdx.x;
    const int half  = lane >> 4;
    const int l16   = lane & 15;
    const int qbase = blockIdx.x * 16;
    const int q     = qbase + l16;

    // Preload this lane's query B-operand values for every K chunk.
    v2f qv[KC];
#pragma unroll
    for (int kc = 0; kc < KC; ++kc) {
        const float* p = &x[q * CIN + kc * 4 + half * 2];
        qv[kc].x = p[0];
        qv[kc].y = p[1];
    }
    const float sqq = sq[q];

    // Register-resident top-20 (dist, idx), replace-max policy.
    float bd[KNN_K];
    int   bi[KNN_K];
#pragma unroll
    for (int t = 0; t < KNN_K; ++t) { bd[t] = FLT_MAX; bi[t] = 0; }
    float curMax = FLT_MAX;

    for (int cbase = 0; cbase < n; cbase += 16) {
        if (cbase + 64 < n)
            __builtin_prefetch(&x[(size_t)(cbase + 64) * CIN], 0, 1);

        v8f acc = v8f_zero();
        const int cRow = cbase + l16;  // A-matrix row owned by this lane
#pragma unroll
        for (int kc = 0; kc < KC; ++kc) {
            const float* p = &x[cRow * CIN + kc * 4 + half * 2];
            v2f a;
            a.x = p[0];
            a.y = p[1];
            acc = wmma_f32(a, qv[kc], acc);  // acc[v] = dot(cand_{half*8+v}, query_{l16})
        }

#pragma unroll
        for (int v = 0; v < 8; ++v) {
            int   c = cbase + half * 8 + v;
            float d = sqq + sq[c] - 2.0f * acc[v];
            if (d < curMax) {
                int   mt = 0;
                float mv = bd[0];
#pragma unroll
                for (int t = 1; t < KNN_K; ++t)
                    if (bd[t] > mv) { mv = bd[t]; mt = t; }
#pragma unroll
                for (int t = 0; t < KNN_K; ++t)
                    if (t == mt) { bd[t] = d; bi[t] = c; }
                float nm = bd[0];
#pragma unroll
                for (int t = 1; t < KNN_K; ++t) nm = fmaxf(nm, bd[t]);
                curMax = nm;
            }
        }
    }

    // Merge the two half-wave lists (40 entries -> smallest 20) via LDS.
    __shared__ float smd[32 * KNN_K];
    __shared__ int   smi[32 * KNN_K];
#pragma unroll
    for (int t = 0; t < KNN_K; ++t) {
        smd[lane * KNN_K + t] = bd[t];
        smi[lane * KNN_K + t] = bi[t];
    }
    __syncthreads();

    if (lane < 16) {
        for (int s = 0; s < KNN_K; ++s) {
            float mv   = FLT_MAX;
            int   mpos = lane * KNN_K;  // safe default
            for (int t = 0; t < 2 * KNN_K; ++t) {
                int ln = (t < KNN_K) ? lane : lane + 16;
                int tt = (t < KNN_K) ? t : t - KNN_K;
                float vv = smd[ln * KNN_K + tt];
                if (vv < mv) { mv = vv; mpos = ln * KNN_K + tt; }
            }
            knn_idx[q * KNN_K + s] = smi[mpos];
            smd[mpos] = FLT_MAX;  // consume
        }
    }
}

// ---------------------------------------------------------------------------
// Fused EdgeConv MLP + segmented max. One wave per 4 points:
// 4 * 20 = 80 edge rows = exactly 5 WMMA M-tiles (no padding waste).
//   feat(2*CIN) -> LDS bufA -> GEMM W1(FPAD x 64) -> leaky -> bufB(64)
//                       bufB -> GEMM W2           -> leaky -> bufA(64)
//                       bufA -> GEMM W3           -> leaky -> bufB, max -> out
// CROW = row stride of x (may exceed CIN for padded layer-0 input).
// W1 must have FPAD = roundup4(2*CIN) rows (zero-padded) -> branchless.
// LDS strides padded (+4 floats) so M-striped A-operand reads hit 16 banks.
// ---------------------------------------------------------------------------
template <int CROW, int CIN, int COUT>
__global__ void __launch_bounds__(32) edge_mlp_kernel(
    const float* __restrict__ x, const int* __restrict__ knn_idx,
    const float* __restrict__ W1, const float* __restrict__ b1,
    const float* __restrict__ W2, const float* __restrict__ b2,
    const float* __restrict__ W3, const float* __restrict__ b3,
    float* __restrict__ out) {
    constexpr int FDIM = 2 * CIN;                 // real feature width (6 or 128)
    constexpr int FPAD = (FDIM + 3) & ~3;         // K padded to mult of 4
    constexpr int KC1  = FPAD / 4;
    constexpr int LDA  = 132;                     // bufA stride (floats)
    constexpr int LDB  = 68;                      // bufB stride (floats)

    __shared__ float bufA[80 * LDA];  // 42240 B
    __shared__ float bufB[80 * LDB];  // 21760 B

    const int lane  = threadIdx.x;
    const int half  = lane >> 4;
    const int l16   = lane & 15;
    const int pbase = blockIdx.x * 4;

    // ---- Stage edge features: row e -> [ x_i , x_j - x_i , 0-pad ] ----
    for (int e = lane; e < 80; e += 32) {
        int p = pbase + e / KNN_K;
        int j = knn_idx[p * KNN_K + (e % KNN_K)];
        const float* xi = x + (size_t)p * CROW;
        const float* xj = x + (size_t)j * CROW;
#pragma unroll 4
        for (int c = 0; c < CIN; ++c) {
            float a = xi[c];
            bufA[e * LDA + c]       = a;
            bufA[e * LDA + CIN + c] = xj[c] - a;
        }
#pragma unroll
        for (int c = FDIM; c < FPAD; ++c) bufA[e * LDA + c] = 0.f;
    }
    __syncthreads();

    // ---- Layer 1: bufA (80 x FPAD) @ W1 (FPAD x 64) -> bufB ----
    for (int mt = 0; mt < 5; ++mt) {
        for (int nt = 0; nt < 4; ++nt) {
            int   colN = nt * 16 + l16;
            float bias = b1[colN];
            v8f   acc  = v8f_zero();
#pragma unroll 4
            for (int kc = 0; kc < KC1; ++kc) {
                int k = kc * 4 + half * 2;
                v2f a, b;
                a.x = bufA[(mt * 16 + l16) * LDA + k];
                a.y = bufA[(mt * 16 + l16) * LDA + k + 1];
                b.x = W1[k * 64 + colN];
                b.y = W1[(k + 1) * 64 + colN];
                acc = wmma_f32(a, b, acc);
            }
#pragma unroll
            for (int v = 0; v < 8; ++v) {
                float h = acc[v] + bias;
                h = (h >= 0.f) ? h : NEG_SLOPE * h;
                bufB[(mt * 16 + half * 8 + v) * LDB + colN] = h;
            }
        }
    }
    __syncthreads();

    // ---- Layer 2: bufB (80 x 64) @ W2 (64 x 64) -> bufA ----
    for (int mt = 0; mt < 5; ++mt) {
        for (int nt = 0; nt < 4; ++nt) {
            int   colN = nt * 16 + l16;
            float bias = b2[colN];
            v8f   acc  = v8f_zero();
#pragma unroll 4
            for (int kc = 0; kc < 16; ++kc) {
                int k = kc * 4 + half * 2;
                v2f a, b;
                a.x = bufB[(mt * 16 + l16) * LDB + k];
                a.y = bufB[(mt * 16 + l16) * LDB + k + 1];
                b.x = W2[k * 64 + colN];
                b.y = W2[(k + 1) * 64 + colN];
                acc = wmma_f32(a, b, acc);
            }
#pragma unroll
            for (int v = 0; v < 8; ++v) {
                float h = acc[v] + bias;
                h = (h >= 0.f) ? h : NEG_SLOPE * h;
                bufA[(mt * 16 + half * 8 + v) * LDA + colN] = h;
            }
        }
    }
    __syncthreads();

    // ---- Layer 3 (+ max over K) in column groups of 64 ----
    constexpr int NG = COUT / 64;  // 1 or 2
    for (int cg = 0; cg < NG; ++cg) {
        for (int mt = 0; mt < 5; ++mt) {
            for (int nt = 0; nt < 4; ++nt) {
                int   colN = cg * 64 + nt * 16 + l16;
                float bias = b3[colN];
                v8f   acc  = v8f_zero();
#pragma unroll 4
                for (int kc = 0; kc < 16; ++kc) {
                    int k = kc * 4 + half * 2;
                    v2f a, b;
                    a.x = bufA[(mt * 16 + l16) * LDA + k];
                    a.y = bufA[(mt * 16 + l16) * LDA + k + 1];
                    b.x = W3[k * COUT + colN];
                    b.y = W3[(k + 1) * COUT + colN];
                    acc = wmma_f32(a, b, acc);
                }
#pragma unroll
                for (int v = 0; v < 8; ++v) {
                    float h = acc[v] + bias;
                    h = (h >= 0.f) ? h : NEG_SLOPE * h;
                    bufB[(mt * 16 + half * 8 + v) * LDB + nt * 16 + l16] = h;
                }
            }
        }
        __syncthreads();

        // Segmented max over each point's 20 edge rows.
        for (int t = lane; t < 4 * 64; t += 32) {
            int   p = t >> 6;
            int   c = t & 63;
            float m = bufB[(p * KNN_K) * LDB + c];
            for (int s = 1; s < KNN_K; ++s)
                m = fmaxf(m, bufB[(p * KNN_K + s) * LDB + c]);
            out[(size_t)(pbase + p) * COUT + cg * 64 + c] = m;
        }
        __syncthreads();
    }
}

// ---------------------------------------------------------------------------
// Host driver. Input order: d_in[0]=pos, then (W,b) x3 per layer group x3.
// Workspace: sq | idx | xb1 | xb2 | xpad (pos padded to 4ch) | W1p (8x64).
// ---------------------------------------------------------------------------
extern "C" void kernel_launch(void* const* d_in, const int* in_sizes, int n_in,
                              void* d_out, int out_size, void* d_ws, size_t ws_size,
                              hipStream_t stream) {
    (void)in_sizes; (void)n_in; (void)out_size; (void)ws_size;

    const float* pos = (const float*)d_in[0];
    const float* W[3][3];
    const float* B[3][3];
    for (int g = 0; g < 3; ++g)
        for (int l = 0; l < 3; ++l) {
            W[g][l] = (const float*)d_in[1 + g * 6 + l * 2];
            B[g][l] = (const float*)d_in[1 + g * 6 + l * 2 + 1];
        }

    char*  ws   = (char*)d_ws;
    float* sqb  = (float*)ws;                                  // 16384 f32      (64 KB)
    int*   idx  = (int*)(ws + 65536);                          // 16384*20 i32   (1.25 MB)
    float* xb1  = (float*)(ws + 65536 + 1310720);              // 16384*64 f32   (4 MB)
    float* xb2  = xb1 + (size_t)NPTS * 64;                     // 16384*64 f32   (4 MB)
    float* xpad = xb2 + (size_t)NPTS * 64;                     // 16384*4  f32   (256 KB)
    float* w1p  = xpad + (size_t)NPTS * 4;                     // 8*64     f32   (2 KB)
    float* outp = (float*)d_out;                               // 16384*128 f32

    const dim3 waveblk(32);
    const dim3 knngrid(NPTS / 16);
    const dim3 mlpgrid(NPTS / 4);

    // ---- Setup: pad pos and W1 of EdgeConv-0 ----
    pad_pos_kernel<<<dim3((NPTS * 4) / 256), dim3(256), 0, stream>>>(pos, xpad, NPTS);
    pad_w1_kernel<<<dim3(2), dim3(256), 0, stream>>>(W[0][0], w1p);

    // ---- EdgeConv 0: xpad (stride 4, C=3) -> xb1 (64) ----
    sqnorm_kernel<4><<<dim3(NPTS / 256), dim3(256), 0, stream>>>(xpad, sqb, NPTS);
    knn_kernel<4><<<knngrid, waveblk, 0, stream>>>(xpad, sqb, idx, NPTS);
    edge_mlp_kernel<4, 3, 64><<<mlpgrid, waveblk, 0, stream>>>(
        xpad, idx, w1p, B[0][0], W[0][1], B[0][1], W[0][2], B[0][2], xb1);

    // ---- EdgeConv 1: xb1 (64) -> xb2 (64) ----
    sqnorm_kernel<64><<<dim3(NPTS / 256), dim3(256), 0, stream>>>(xb1, sqb, NPTS);
    knn_kernel<64><<<knngrid, waveblk, 0, stream>>>(xb1, sqb, idx, NPTS);
    edge_mlp_kernel<64, 64, 64><<<mlpgrid, waveblk, 0, stream>>>(
        xb1, idx, W[1][0], B[1][0], W[1][1], B[1][1], W[1][2], B[1][2], xb2);

    // ---- EdgeConv 2: xb2 (64) -> d_out (128) ----
    sqnorm_kernel<64><<<dim3(NPTS / 256), dim3(256), 0, stream>>>(xb2, sqb, NPTS);
    knn_kernel<64><<<knngrid, waveblk, 0, stream>>>(xb2, sqb, idx, NPTS);
    edge_mlp_kernel<64, 64, 128><<<mlpgrid, waveblk, 0, stream>>>(
        xb2, idx, W[2][0], B[2][0], W[2][1], B[2][1], W[2][2], B[2][2], outp);
}